// HGTD_55284819034840
// MI455X (gfx1250) — compile-verified
//
#include <hip/hip_runtime.h>
#include <hip/hip_bf16.h>

#define DIM1 200
#define DIM2 200
#define DIM3 500
#define RANK 30
#define LAMDA 0.001f

#define KP 32            // K padded 30 -> 32 (zero-filled)
#define NROWP 208        // N rows padded 200 -> 13*16 (zero rows)
#define TROWP 512        // T rows padded 500 -> 32*16 (zero rows)

typedef __attribute__((ext_vector_type(2))) float v2f;
typedef __attribute__((ext_vector_type(8))) float v8f;

// ---- workspace layout (float offsets) ----
// 6 accumulators: [0]=loss1_sq [1]=|N|^2 [2]=|D|^2 [3]=|T|^2 [4]=loss3_sq [5]=loss4_sq
#define WS_ACC   0
#define WS_NP    6
#define WS_DP    (WS_NP  + NROWP*KP)
#define WS_TP    (WS_DP  + DIM2*KP)
#define WS_NB    (WS_TP  + TROWP*KP)
#define WS_DB    (WS_NB  + DIM1)
#define WS_TBP   (WS_DB  + DIM2)
#define WS_SUMD  (WS_TBP + TROWP)
#define WS_NNT   (WS_SUMD + DIM2)
#define WS_TOTAL (WS_NNT + DIM1*DIM1)    // ~70558 floats (~282 KB)

__device__ __forceinline__ float softplus_f(float x) { return log1pf(expf(x)); }

__device__ __forceinline__ float wave_sum(float v) {
#pragma unroll
    for (int o = 16; o > 0; o >>= 1) v += __shfl_xor(v, o, 32);
    return v;
}

// ---------------- kernel 1: sample factors into PADDED layouts + norms + sumD ----------------
__global__ void sample_kernel(const float* Nmu, const float* Nrho, const float* epsN,
                              const float* Dmu, const float* Drho, const float* epsD,
                              const float* Tmu, const float* Trho, const float* epsT,
                              const float* Nbmu, const float* Nbrho, const float* epsNb,
                              const float* Dbmu, const float* Dbrho, const float* epsDb,
                              const float* Tbmu, const float* Tbrho, const float* epsTb,
                              float* ws) {
    const int nNP = NROWP * KP;          // 6656
    const int nDP = DIM2 * KP;           // 6400
    const int nTP = TROWP * KP;          // 16384
    int i = blockIdx.x * blockDim.x + threadIdx.x;
    float accN = 0.f, accD = 0.f, accT = 0.f;
    if (i < nNP) {
        const int row = i >> 5, k = i & 31;
        float w = 0.f;
        if (row < DIM1 && k < RANK) {
            const int j = row * RANK + k;
            w = Nmu[j] + softplus_f(Nrho[j]) * epsN[j];
        }
        ws[WS_NP + i] = w; accN = w * w;
    } else if (i < nNP + nDP) {
        const int p = i - nNP, row = p >> 5, k = p & 31;
        float w = 0.f;
        if (k < RANK) {
            const int j = row * RANK + k;
            w = Dmu[j] + softplus_f(Drho[j]) * epsD[j];
        }
        ws[WS_DP + p] = w; accD = w * w;
    } else if (i < nNP + nDP + nTP) {
        const int p = i - nNP - nDP, row = p >> 5, k = p & 31;
        float w = 0.f;
        if (row < DIM3 && k < RANK) {
            const int j = row * RANK + k;
            w = Tmu[j] + softplus_f(Trho[j]) * epsT[j];
        }
        ws[WS_TP + p] = w; accT = w * w;
    } else if (i < nNP + nDP + nTP + DIM1) {
        const int j = i - nNP - nDP - nTP;
        ws[WS_NB + j] = Nbmu[j] + softplus_f(Nbrho[j]) * epsNb[j];
    } else if (i < nNP + nDP + nTP + DIM1 + DIM2) {
        const int j = i - nNP - nDP - nTP - DIM1;
        ws[WS_DB + j] = Dbmu[j] + softplus_f(Dbrho[j]) * epsDb[j];
    } else if (i < nNP + nDP + nTP + DIM1 + DIM2 + TROWP) {
        const int j = i - nNP - nDP - nTP - DIM1 - DIM2;
        ws[WS_TBP + j] = (j < DIM3) ? (Tbmu[j] + softplus_f(Tbrho[j]) * epsTb[j]) : 0.f;
    } else if (i < nNP + nDP + nTP + DIM1 + DIM2 + TROWP + DIM2) {
        // sumD[d] recomputed directly from inputs (no cross-block dependency)
        const int d = i - nNP - nDP - nTP - DIM1 - DIM2 - TROWP;
        float s = 0.f;
#pragma unroll
        for (int r = 0; r < RANK; ++r) {
            const int j = d * RANK + r;
            s += Dmu[j] + softplus_f(Drho[j]) * epsD[j];
        }
        ws[WS_SUMD + d] = s;
    }
    accN = wave_sum(accN); accD = wave_sum(accD); accT = wave_sum(accT);
    if ((threadIdx.x & 31) == 0) {
        if (accN != 0.f) atomicAdd(&ws[1], accN);
        if (accD != 0.f) atomicAdd(&ws[2], accD);
        if (accT != 0.f) atomicAdd(&ws[3], accT);
    }
}

// ---------------- kernel 2: NNt = N @ N^T via WMMA + loss3 partial ----------------
// one wave per 16x16 tile, 13x13 tiles over padded 208x208; K = 32 (padded), no guards in loop
__global__ void nnt_kernel(const float* __restrict__ hs, float* ws) {
    const float* Nf = ws + WS_NP;
    float* nnt = ws + WS_NNT;
    const int lane = threadIdx.x;            // 32 threads / block
    const int ntiles = NROWP / 16;           // 13
    const int it = blockIdx.x / ntiles;
    const int jt = blockIdx.x % ntiles;
    const int half = lane >> 4;
    const int l16 = lane & 15;
    const int rowA = it * 16 + l16;          // < 208, always valid in padded array
    const int colB = jt * 16 + l16;

    // batch all fragment loads (unconditional, 8B-aligned b64), then WMMA chain
    v2f aF[8], bF[8];
#pragma unroll
    for (int kk = 0; kk < 8; ++kk) {
        const int ka = kk * 4 + half * 2;    // lanes 0-15: K=k0,k0+1 ; lanes 16-31: K=k0+2,k0+3
        aF[kk] = *(const v2f*)(Nf + rowA * KP + ka);
        bF[kk] = *(const v2f*)(Nf + colB * KP + ka);
    }
    v8f c = {};
#pragma unroll
    for (int kk = 0; kk < 8; ++kk)
        c = __builtin_amdgcn_wmma_f32_16x16x4_f32(false, aF[kk], false, bF[kk],
                                                  (short)0, c, false, false);

    const int jj = jt * 16 + l16;
    float acc = 0.f;
    if (it < 12 && jt < 12) {
        // interior tile: all rows/cols < 192 < 200 -> fully unguarded epilogue
#pragma unroll
        for (int j = 0; j < 8; ++j) {        // C/D: VGPR j -> row j (lanes 0-15) / j+8 (16-31)
            const int i = it * 16 + j + half * 8;
            const float v = c[j];
            nnt[i * DIM1 + jj] = v;
            const float d = hs[i * DIM1 + jj] - v;
            acc += d * d;
        }
    } else {
        // edge tile: per-lane guards
#pragma unroll
        for (int j = 0; j < 8; ++j) {
            const int i = it * 16 + j + half * 8;
            if (i < DIM1 && jj < DIM1) {
                const float v = c[j];
                nnt[i * DIM1 + jj] = v;
                const float d = hs[i * DIM1 + jj] - v;
                acc += d * d;
            }
        }
    }
    acc = wave_sum(acc);
    if (lane == 0) atomicAdd(&ws[4], acc);
}

// ---------------- kernel 3: loss4 streaming reduction over 200^3 ----------------
__global__ void loss4_kernel(const float* __restrict__ ht, float* ws) {
    const float* nnt  = ws + WS_NNT;
    const float* sumD = ws + WS_SUMD;
    const int total = DIM1 * DIM1 * DIM2;   // 8,000,000 (fits int32)
    float acc = 0.f;
    const int stride = gridDim.x * blockDim.x;
    for (int i = blockIdx.x * blockDim.x + threadIdx.x; i < total; i += stride) {
        const int d  = i % DIM2;            // magic-multiply, 32-bit
        const int nm = i / DIM2;
        const float diff = __builtin_nontemporal_load(&ht[i]) - nnt[nm] * sumD[d];
        acc += diff * diff;
    }
    acc = wave_sum(acc);
    if ((threadIdx.x & 31) == 0) atomicAdd(&ws[5], acc);
}

// ---------------- kernel 4: pred GEMM (40000x512pad, K=32pad) via WMMA + loss1 ----------------
// W[m,r] = N[m/200, r] * D[m%200, r]; pred = W @ T^T. One wave per 16x16 tile.
// 2500 M-tiles x 32 T-tiles = 80000 waves, 4 waves/block -> 20000 blocks (exact, no tail).
__global__ void pred_kernel(const float* __restrict__ flow,
                            const unsigned char* __restrict__ mask,
                            float* ws, float* __restrict__ pred) {
    const float* Nf = ws + WS_NP;
    const float* Df = ws + WS_DP;
    const float* Tf = ws + WS_TP;
    const float* Nb = ws + WS_NB;
    const float* Db = ws + WS_DB;
    const float* Tb = ws + WS_TBP;

    const int lane = threadIdx.x & 31;
    const int wave = threadIdx.x >> 5;
    const int tile = blockIdx.x * (blockDim.x >> 5) + wave;
    const int TT = TROWP / 16;               // 32 column tiles (last has t>=500 zero-padded)
    const int mt = tile / TT;                // 0..2499
    const int tt = tile % TT;
    const int half = lane >> 4;
    const int l16 = lane & 15;

    const int rowA = mt * 16 + l16;          // always < 40000
    const int na = rowA / DIM2;
    const int da = rowA % DIM2;
    const int t = tt * 16 + l16;             // < 512, always valid in padded T

    // batch all fragment loads (no guards: padding supplies zeros), then WMMA chain
    v2f aF[8], bF[8];
#pragma unroll
    for (int kk = 0; kk < 8; ++kk) {
        const int ka = kk * 4 + half * 2;
        const v2f n2 = *(const v2f*)(Nf + na * KP + ka);
        const v2f d2 = *(const v2f*)(Df + da * KP + ka);
        aF[kk] = n2 * d2;
        bF[kk] = *(const v2f*)(Tf + t * KP + ka);
    }
    v8f c = {};
#pragma unroll
    for (int kk = 0; kk < 8; ++kk)
        c = __builtin_amdgcn_wmma_f32_16x16x4_f32(false, aF[kk], false, bF[kk],
                                                  (short)0, c, false, false);

    const float tb = Tb[t];                  // zero in padding
    // one div/mod per wave; per-j row decomposition is a branchless wrap
    const int rowBase = mt * 16;
    const int n0 = rowBase / DIM2;
    const int d0 = rowBase % DIM2;
    float acc = 0.f;

    if (tt < 31) {
        // full tile: t in [tt*16, tt*16+15] all < 500 -> unguarded, branchless epilogue
#pragma unroll
        for (int j = 0; j < 8; ++j) {
            const int off = j + half * 8;
            const int wrap = (d0 + off >= DIM2) ? 1 : 0;
            const int dd = d0 + off - wrap * DIM2;
            const int nn = n0 + wrap;
            const float v = c[j] + Nb[nn] + Db[dd] + tb;
            const int row = rowBase + off;
            const long idx = (long)row * DIM3 + t;
            __builtin_nontemporal_store(v, &pred[idx]);            // 80 MB stream
            const float fl = __builtin_nontemporal_load(&flow[idx]);
            const float mk = __builtin_nontemporal_load(&mask[idx]) ? 0.f : 1.f;
            const float df = (v - fl) * mk;                        // branchless masked diff
            acc += df * df;
        }
    } else {
        // edge tile: lanes with t >= 500 are padding
        const bool tOK = t < DIM3;
#pragma unroll
        for (int j = 0; j < 8; ++j) {
            const int off = j + half * 8;
            const int wrap = (d0 + off >= DIM2) ? 1 : 0;
            const int dd = d0 + off - wrap * DIM2;
            const int nn = n0 + wrap;
            const float v = c[j] + Nb[nn] + Db[dd] + tb;
            if (tOK) {
                const int row = rowBase + off;
                const long idx = (long)row * DIM3 + t;
                __builtin_nontemporal_store(v, &pred[idx]);
                const float fl = __builtin_nontemporal_load(&flow[idx]);
                const float mk = __builtin_nontemporal_load(&mask[idx]) ? 0.f : 1.f;
                const float df = (v - fl) * mk;
                acc += df * df;
            }
        }
    }
    acc = wave_sum(acc);
    if (lane == 0) atomicAdd(&ws[0], acc);
}

// ---------------- kernel 5: combine losses ----------------
__global__ void finalize_kernel(const float* ws, float* out) {
    if (threadIdx.x == 0 && blockIdx.x == 0) {
        const float loss = sqrtf(ws[0])
                         + LAMDA * (sqrtf(ws[1]) + sqrtf(ws[2]) + sqrtf(ws[3]))
                         + LAMDA * sqrtf(ws[4])
                         + LAMDA * sqrtf(ws[5]);
        out[0] = loss;
    }
}

extern "C" void kernel_launch(void* const* d_in, const int* in_sizes, int n_in,
                              void* d_out, int out_size, void* d_ws, size_t ws_size,
                              hipStream_t stream) {
    const float* flow = (const float*)d_in[0];
    const unsigned char* mask = (const unsigned char*)d_in[1];   // jnp bool
    const float* hs = (const float*)d_in[2];
    const float* ht = (const float*)d_in[3];
    const float* Nmu  = (const float*)d_in[4];
    const float* Nrho = (const float*)d_in[5];
    const float* epsN = (const float*)d_in[6];
    const float* Dmu  = (const float*)d_in[7];
    const float* Drho = (const float*)d_in[8];
    const float* epsD = (const float*)d_in[9];
    const float* Tmu  = (const float*)d_in[10];
    const float* Trho = (const float*)d_in[11];
    const float* epsT = (const float*)d_in[12];
    const float* Nbmu  = (const float*)d_in[13];
    const float* Nbrho = (const float*)d_in[14];
    const float* epsNb = (const float*)d_in[15];
    const float* Dbmu  = (const float*)d_in[16];
    const float* Dbrho = (const float*)d_in[17];
    const float* epsDb = (const float*)d_in[18];
    const float* Tbmu  = (const float*)d_in[19];
    const float* Tbrho = (const float*)d_in[20];
    const float* epsTb = (const float*)d_in[21];

    float* out = (float*)d_out;          // out[0]=loss, out[1..]=pred (20M f32)
    float* ws = (float*)d_ws;

    // zero the 6 loss accumulators (graph-capture safe)
    hipMemsetAsync(ws, 0, 6 * sizeof(float), stream);

    const int nSample = NROWP*KP + DIM2*KP + TROWP*KP + DIM1 + DIM2 + TROWP + DIM2; // 30552
    sample_kernel<<<(nSample + 255) / 256, 256, 0, stream>>>(
        Nmu, Nrho, epsN, Dmu, Drho, epsD, Tmu, Trho, epsT,
        Nbmu, Nbrho, epsNb, Dbmu, Dbrho, epsDb, Tbmu, Tbrho, epsTb, ws);

    const int ntiles = NROWP / 16;                       // 13
    nnt_kernel<<<ntiles * ntiles, 32, 0, stream>>>(hs, ws);

    loss4_kernel<<<4096, 256, 0, stream>>>(ht, ws);

    const int TT = TROWP / 16;                           // 32
    const int nWaves = (DIM1 * DIM2 / 16) * TT;          // 2500*32 = 80000
    pred_kernel<<<nWaves / 4, 128, 0, stream>>>(flow, mask, ws, out + 1);

    finalize_kernel<<<1, 1, 0, stream>>>(ws, out);
}